// LocalDynamicGraph_56538949484665
// MI455X (gfx1250) — compile-verified
//
#include <hip/hip_runtime.h>
#include <stdint.h>

#ifndef __has_builtin
#define __has_builtin(x) 0
#endif

// ---- CDNA5 async global<->LDS DMA availability probes -----------------------
#define HAVE_ASYNC_LD (__has_builtin(__builtin_amdgcn_global_load_async_to_lds_b128) && \
                       __has_builtin(__builtin_amdgcn_global_load_async_to_lds_b32)  && \
                       __has_builtin(__builtin_amdgcn_s_wait_asynccnt))
#define HAVE_ASYNC_ST (__has_builtin(__builtin_amdgcn_global_store_async_from_lds_b128) && \
                       __has_builtin(__builtin_amdgcn_s_wait_asynccnt))

#if HAVE_ASYNC_LD
#pragma message("CDNA5 probe: global_load_async_to_lds builtins AVAILABLE")
#else
#pragma message("CDNA5 probe: global_load_async_to_lds builtins NOT available -> fallback loads")
#endif
#if HAVE_ASYNC_ST
#pragma message("CDNA5 probe: global_store_async_from_lds builtins AVAILABLE")
#else
#pragma message("CDNA5 probe: global_store_async_from_lds builtins NOT available -> fallback stores")
#endif

// Problem sizes (fixed by the reference).
static constexpr int kB   = 8;
static constexpr int kN   = 16384;           // 1 << 14
static constexpr int kC   = 64;
static constexpr int kK   = 20;
static constexpr int TPB  = 256;             // 8 waves (wave32)
static constexpr int TILE = 2 * kC * kK;     // 2560 floats per point (128 x 20)
static constexpr int GVEC = kK * (kC / 4);   // 320 float4 gathers per point

// gfx12+ cache-policy immediates: TH in bits [2:0]; TH_NT = 1.
// NT on the 1.34 GB output stream (and the read-once idx stream) keeps the
// heavily-reused x array (33.5 MB, ~20x gather reuse) resident in the 192 MB L2.
static constexpr int CPOL_RT = 0;
static constexpr int CPOL_NT = 1;

// Types matching the builtin signatures (from clang diagnostics):
//   arg0: addrspace(1) T*   (printed '__device__ ...')
//   arg1: addrspace(3) T*
// with T = int for b32 and T = int __attribute__((vector_size(16))) for b128.
typedef int v4i __attribute__((vector_size(16)));
typedef __attribute__((address_space(1))) int g_i32_t;
typedef __attribute__((address_space(1))) v4i g_v4i_t;
typedef __attribute__((address_space(3))) int l_i32_t;
typedef __attribute__((address_space(3))) v4i l_v4i_t;

// Flat->global: bit-identical 64-bit address.
__device__ __forceinline__ g_i32_t* glb_i32(const void* p) {
    return (g_i32_t*)(uintptr_t)p;
}
__device__ __forceinline__ g_v4i_t* glb_v4i(const void* p) {
    return (g_v4i_t*)(uintptr_t)p;
}
// Flat->LDS: low 32 bits of a flat shared address are the LDS byte offset
// (aperture check uses addr[63:32]).
__device__ __forceinline__ l_i32_t* lds_i32(const void* p) {
    return (l_i32_t*)(uint32_t)(uintptr_t)p;
}
__device__ __forceinline__ l_v4i_t* lds_v4i(const void* p) {
    return (l_v4i_t*)(uint32_t)(uintptr_t)p;
}

__global__ __launch_bounds__(TPB) void
edge_feature_kernel(const float* __restrict__ x,   // (B, N, C)
                    const int*   __restrict__ idx, // (B, N, K)
                    float*       __restrict__ out) // (B, N, 2C, K)
{
    __shared__ __align__(16) float s_tile[TILE];     // transposed tile [c2 * kK + k]
    __shared__ __align__(16) float s_nbr[kK * kC];   // raw gathered neighbor rows
    __shared__ __align__(16) float s_center[kC];
    __shared__ __align__(16) int   s_idx[kK];

    const int point = blockIdx.x;        // b * kN + n
    const int b     = point >> 14;       // kN == 1 << 14
    const int t     = threadIdx.x;

    const float* xb = x + (size_t)b * kN * kC;   // batch base for gathers
    const float* xp = x + (size_t)point * kC;    // center row
    const int*   ip = idx + (size_t)point * kK;

#if HAVE_ASYNC_LD
    // Phase 0: DMA indices (20 x b32, streaming/NT) and center row (16 x b128)
    // into LDS.
    if (t < kK) {
        __builtin_amdgcn_global_load_async_to_lds_b32(
            glb_i32(ip + t), lds_i32(s_idx + t), 0, CPOL_NT);
    } else if (t >= 32 && t < 32 + kC / 4) {
        const int c = (t - 32) * 4;
        __builtin_amdgcn_global_load_async_to_lds_b128(
            glb_v4i(xp + c), lds_v4i(s_center + c), 0, CPOL_RT);
    }
    __builtin_amdgcn_s_wait_asynccnt(0);
    __syncthreads();

    // Phase 1: DMA-gather the 20 neighbor rows (320 x b128, coalesced per row).
    // RT policy: x has ~20x reuse and lives in L2.
    for (int e = t; e < GVEC; e += TPB) {
        const int k = e >> 4;
        const int c = (e & 15) * 4;
        __builtin_amdgcn_global_load_async_to_lds_b128(
            glb_v4i(xb + (size_t)s_idx[k] * kC + c),
            lds_v4i(s_nbr + k * kC + c), 0, CPOL_RT);
    }
    __builtin_amdgcn_s_wait_asynccnt(0);
    __syncthreads();
#else
    if (t < kK) s_idx[t] = ip[t];
    if (t >= 32 && t < 32 + kC) s_center[t - 32] = xp[t - 32];
    __syncthreads();
    for (int e = t; e < GVEC; e += TPB) {
        const int k = e >> 4;
        const int c = (e & 15) * 4;
        *(float4*)(s_nbr + k * kC + c) =
            *(const float4*)(xb + (size_t)s_idx[k] * kC + c);
    }
    __syncthreads();
#endif

    // Phase 2: edge features, written directly in transposed [2C][K] layout.
    for (int e = t; e < GVEC; e += TPB) {
        const int k = e >> 4;
        const int c = (e & 15) * 4;
        const float4 v = *(const float4*)(s_nbr + k * kC + c);
        const float c0 = s_center[c + 0], c1 = s_center[c + 1];
        const float c2 = s_center[c + 2], c3 = s_center[c + 3];
        s_tile[(c + 0) * kK + k] = v.x - c0;
        s_tile[(c + 1) * kK + k] = v.y - c1;
        s_tile[(c + 2) * kK + k] = v.z - c2;
        s_tile[(c + 3) * kK + k] = v.w - c3;
        s_tile[(kC + c + 0) * kK + k] = c0;
        s_tile[(kC + c + 1) * kK + k] = c1;
        s_tile[(kC + c + 2) * kK + k] = c2;
        s_tile[(kC + c + 3) * kK + k] = c3;
    }
    __syncthreads();

    // Phase 3: stream the tile out as contiguous 128-bit stores (non-temporal:
    // pure streaming output, never re-read; don't let it evict x from L2).
    float* op = out + (size_t)point * TILE;
#if HAVE_ASYNC_ST
    for (int e = t; e < TILE / 4; e += TPB) {
        __builtin_amdgcn_global_store_async_from_lds_b128(
            glb_v4i(op + e * 4), lds_v4i(s_tile + e * 4), 0, CPOL_NT);
    }
    __builtin_amdgcn_s_wait_asynccnt(0);
#else
    for (int e = t; e < TILE / 4; e += TPB) {
        *(float4*)(op + e * 4) = *(const float4*)(s_tile + e * 4);
    }
#endif
}

extern "C" void kernel_launch(void* const* d_in, const int* in_sizes, int n_in,
                              void* d_out, int out_size, void* d_ws, size_t ws_size,
                              hipStream_t stream) {
    (void)in_sizes; (void)n_in; (void)out_size; (void)d_ws; (void)ws_size;
    const float* x   = (const float*)d_in[0];
    const int*   idx = (const int*)d_in[1];
    float*       out = (float*)d_out;

    edge_feature_kernel<<<dim3(kB * kN), dim3(TPB), 0, stream>>>(x, idx, out);
}